// HypernymVisual_acc_10213432230335
// MI455X (gfx1250) — compile-verified
//
#include <hip/hip_runtime.h>

typedef __attribute__((ext_vector_type(2))) float v2f;
typedef __attribute__((ext_vector_type(8))) float v8f;

#define B_ 512
#define D_ 4096
#define E_ 512
#define N_ 1000

// ---------------------------------------------------------------------------
// Kernel 1: BatchNorm (training-mode batch stats) fused with normalization.
// One thread per feature column d; two coalesced passes over the 512 rows.
// 16 MB total traffic -> <1 us at 23.3 TB/s; not worth more machinery.
// ---------------------------------------------------------------------------
__global__ __launch_bounds__(256) void bn_normalize_kernel(
    const float* __restrict__ vf, const float* __restrict__ gamma,
    const float* __restrict__ beta, float* __restrict__ vfn)
{
    const int d = blockIdx.x * 256 + threadIdx.x;   // d < 4096 by grid shape
    float s = 0.f, s2 = 0.f;
    for (int i = 0; i < B_; ++i) {
        float v = vf[i * D_ + d];
        s += v;
        s2 = fmaf(v, v, s2);
    }
    const float inv = 1.0f / (float)B_;
    const float mean = s * inv;
    const float var  = fmaf(-mean, mean, s2 * inv);   // E[x^2]-mean^2 (biased)
    const float scale = gamma[d] * rsqrtf(var + 1e-5f);
    const float shift = fmaf(-mean, scale, beta[d]);
    for (int i = 0; i < B_; ++i) {
        vfn[i * D_ + d] = fmaf(vf[i * D_ + d], scale, shift);
    }
}

// ---------------------------------------------------------------------------
// Kernel 2: emb = vf_n @ W^T + b via V_WMMA_F32_16X16X4_F32 (full-f32 path).
// Block = 256 threads = 8 waves, 64(m) x 32(n) tile; waves 4(m) x 2(n), each
// owning a 16x16 f32 accumulator. K staged through LDS in 32-chunks via
// b128 loads/stores; row stride 36 floats keeps rows 16B-aligned and the
// lane-strided fragment reads conflict-light (gcd(36,64)=4, 9 invertible
// mod 16 -> 16 distinct banks across the 16-lane half).
// No divergence -> EXEC all-1s as WMMA requires.
// ---------------------------------------------------------------------------
#define KC 32
#define SAS (KC + 4)   // 36 floats = 144 B row stride (16B aligned)
__global__ __launch_bounds__(256) void emb_gemm_wmma_kernel(
    const float* __restrict__ vfn, const float* __restrict__ W,
    const float* __restrict__ bias, float* __restrict__ emb)
{
    __shared__ float sA[64][SAS];
    __shared__ float sB[32][SAS];

    const int t    = threadIdx.x;
    const int lane = t & 31;
    const int wid  = t >> 5;         // 0..7
    const int wm   = wid & 3;        // wave's 16-row group within 64
    const int wn   = wid >> 2;       // wave's 16-col group within 32
    const int l16  = lane & 15;
    const int half = lane >> 4;      // 0: K=0,1  1: K=2,3 (A/B VGPR layout)
    const int m0   = blockIdx.x * 64;
    const int n0   = blockIdx.y * 32;

    v8f acc = {};

    for (int k0 = 0; k0 < D_; k0 += KC) {
        // Stage A: 64 x 32 chunk of vf_n, 2 float4 per thread (b128 path)
        #pragma unroll
        for (int u = 0; u < 2; ++u) {
            int f  = u * 256 + t;
            int r  = f >> 3;          // 8 float4 per row
            int c4 = f & 7;
            *reinterpret_cast<float4*>(&sA[r][c4 * 4]) =
                *reinterpret_cast<const float4*>(&vfn[(m0 + r) * D_ + k0 + c4 * 4]);
        }
        // Stage B: 32 x 32 chunk of W (B[k][n] = W[n][k]), 1 float4 per thread
        {
            int r  = t >> 3;
            int c4 = t & 7;
            *reinterpret_cast<float4*>(&sB[r][c4 * 4]) =
                *reinterpret_cast<const float4*>(&W[(n0 + r) * D_ + k0 + c4 * 4]);
        }
        // Cover first-touch latency of the next chunk (global_prefetch_b8).
        if (k0 + KC < D_) {
            __builtin_prefetch(&vfn[(m0 + (t & 63)) * D_ + k0 + KC], 0, 3);
            __builtin_prefetch(&W[(n0 + (t & 31)) * D_ + k0 + KC], 0, 3);
        }
        __syncthreads();

        #pragma unroll
        for (int kk = 0; kk < KC; kk += 4) {
            // A 16x4: lanes 0-15 hold (M=l16, K=kk+0/1), lanes 16-31 K=kk+2/3
            // B 4x16: symmetric striping (N across lanes, K split by half)
            const int kc = kk + half * 2;
            v2f a, b2;
            a.x  = sA[wm * 16 + l16][kc];
            a.y  = sA[wm * 16 + l16][kc + 1];
            b2.x = sB[wn * 16 + l16][kc];
            b2.y = sB[wn * 16 + l16][kc + 1];
            acc = __builtin_amdgcn_wmma_f32_16x16x4_f32(
                /*neg_a=*/false, a, /*neg_b=*/false, b2,
                /*c_mod=*/(short)0, acc, /*reuse_a=*/false, /*reuse_b=*/false);
        }
        __syncthreads();
    }

    // C/D layout: VGPR r, lanes 0-15 -> (M=r, N=lane); lanes 16-31 -> (M=r+8)
    const int ncol  = n0 + wn * 16 + l16;
    const int mbase = m0 + wm * 16 + half * 8;
    const float bv  = bias[ncol];
    #pragma unroll
    for (int r = 0; r < 8; ++r) {
        emb[(mbase + r) * E_ + ncol] = acc[r] + bv;
    }
}

// ---------------------------------------------------------------------------
// Kernel 3: p_e[i] = sum_j relu(p_wfs[i,j] - emb[i,j])^2  (one block per row)
// ---------------------------------------------------------------------------
__global__ __launch_bounds__(256) void pe_kernel(
    const float* __restrict__ p_wfs, const float* __restrict__ emb,
    float* __restrict__ p_e)
{
    __shared__ float red[256];
    const int i = blockIdx.x;
    const int t = threadIdx.x;
    float s = 0.f;
    for (int j = t; j < E_; j += 256) {
        float d = p_wfs[i * E_ + j] - emb[i * E_ + j];
        d = fmaxf(d, 0.f);
        s = fmaf(d, d, s);
    }
    red[t] = s;
    __syncthreads();
    for (int off = 128; off > 0; off >>= 1) {
        if (t < off) red[t] += red[t + off];
        __syncthreads();
    }
    if (t == 0) p_e[i] = red[0];
}

// ---------------------------------------------------------------------------
// Kernel 4 (roofline-dominant): n_e[i,k] = sum_j relu(n_wfs[k,j]-emb[i,j])^2.
// 32(i) x 64(k) tile per block, E staged through LDS in chunks of 64 via
// b128 loads/stores. Row stride 68 floats: rows stay 16B-aligned; the
// k-indexed sN reads are at worst 2-way conflicted (68 mod 64 = 4, lanes
// tx / tx+16 alias) while sE reads are pure hardware broadcasts.
// Register blocking: 8 accumulators per thread (4 i-rows x 2 k-cols)
// -> 6 DS reads per 24 VALU ops, v_fma/v_max bound as it should be.
// Also broadcasts p_e into the first half of the output (p_e_stack).
// ---------------------------------------------------------------------------
#define JC 64
#define SNS (JC + 4)   // 68 floats = 272 B row stride (16B aligned)
__global__ __launch_bounds__(256) void ne_kernel(
    const float* __restrict__ n_wfs, const float* __restrict__ emb,
    const float* __restrict__ p_e, float* __restrict__ out)
{
    __shared__ float sE[32][SNS];
    __shared__ float sN[64][SNS];

    const int t  = threadIdx.x;
    const int tx = t & 31;     // local k (and k+32)
    const int ty = t >> 5;     // 0..7 -> 4 local i rows each
    const int i0 = blockIdx.x * 32;
    const int k0 = blockIdx.y * 64;

    float acc[4][2] = {};

    for (int j0 = 0; j0 < E_; j0 += JC) {
        // Stage emb tile 32 x 64: 2 float4 per thread, coalesced b128
        #pragma unroll
        for (int u = 0; u < 2; ++u) {
            int f  = u * 256 + t;
            int r  = f >> 4;          // 16 float4 per row
            int c4 = f & 15;
            *reinterpret_cast<float4*>(&sE[r][c4 * 4]) =
                *reinterpret_cast<const float4*>(&emb[(i0 + r) * E_ + j0 + c4 * 4]);
        }
        // Stage n_wfs tile 64 x 64: 4 float4 per thread, guarded for k>=1000
        #pragma unroll
        for (int u = 0; u < 4; ++u) {
            int f  = u * 256 + t;
            int r  = f >> 4;
            int c4 = f & 15;
            int k  = k0 + r;
            float4 v = make_float4(0.f, 0.f, 0.f, 0.f);
            if (k < N_)
                v = *reinterpret_cast<const float4*>(&n_wfs[k * E_ + j0 + c4 * 4]);
            *reinterpret_cast<float4*>(&sN[r][c4 * 4]) = v;
        }
        __syncthreads();

        #pragma unroll 8
        for (int j = 0; j < JC; ++j) {
            const float nv0 = sN[tx][j];
            const float nv1 = sN[tx + 32][j];
            #pragma unroll
            for (int r = 0; r < 4; ++r) {
                const float ev = sE[ty * 4 + r][j];
                float d0 = fmaxf(nv0 - ev, 0.f);
                float d1 = fmaxf(nv1 - ev, 0.f);
                acc[r][0] = fmaf(d0, d0, acc[r][0]);
                acc[r][1] = fmaf(d1, d1, acc[r][1]);
            }
        }
        __syncthreads();
    }

    #pragma unroll
    for (int r = 0; r < 4; ++r) {
        const int i = i0 + ty * 4 + r;
        const float pv = p_e[i];
        #pragma unroll
        for (int kk = 0; kk < 2; ++kk) {
            const int k = k0 + tx + kk * 32;
            if (k < N_) {
                const long idx = (long)i * N_ + k;
                out[idx] = pv;                        // p_e_stack (first B*N)
                out[(long)B_ * N_ + idx] = acc[r][kk]; // n_e_stack (second B*N)
            }
        }
    }
}

// ---------------------------------------------------------------------------
// Launch. Workspace layout (floats): vfn[B*D] | emb[B*E] | p_e[B]  (~9.4 MB)
// ---------------------------------------------------------------------------
extern "C" void kernel_launch(void* const* d_in, const int* in_sizes, int n_in,
                              void* d_out, int out_size, void* d_ws, size_t ws_size,
                              hipStream_t stream) {
    (void)in_sizes; (void)n_in; (void)out_size; (void)ws_size;
    const float* vf    = (const float*)d_in[0];
    const float* p_wfs = (const float*)d_in[1];
    const float* n_wfs = (const float*)d_in[2];
    const float* gamma = (const float*)d_in[3];
    const float* beta  = (const float*)d_in[4];
    const float* W     = (const float*)d_in[5];
    const float* b     = (const float*)d_in[6];
    float* out = (float*)d_out;

    float* ws  = (float*)d_ws;
    float* vfn = ws;                            // B*D
    float* emb = vfn + (size_t)B_ * D_;         // B*E
    float* pe  = emb + (size_t)B_ * E_;         // B

    bn_normalize_kernel<<<D_ / 256, 256, 0, stream>>>(vf, gamma, beta, vfn);
    emb_gemm_wmma_kernel<<<dim3(B_ / 64, E_ / 32), 256, 0, stream>>>(vfn, W, b, emb);
    pe_kernel<<<B_, 256, 0, stream>>>(p_wfs, emb, pe);
    ne_kernel<<<dim3(B_ / 32, (N_ + 63) / 64), 256, 0, stream>>>(n_wfs, emb, pe, out);
}